// RGCNClassifier_88648124990830
// MI455X (gfx1250) — compile-verified
//
#include <hip/hip_runtime.h>

#define N_NODES  100000
#define N_EDGES  1000000
#define N_GRAPHS 512
#define HID      64
#define NREL     3

typedef __attribute__((ext_vector_type(2))) float v2f;
typedef __attribute__((ext_vector_type(8))) float v8f;

// ---------------- utility kernels ----------------

__global__ void zero_u32(unsigned int* p, int n) {
  int i = blockIdx.x * blockDim.x + threadIdx.x;
  if (i < n) p[i] = 0u;
}

__global__ void count_edges(const int* __restrict__ etype,
                            const int* __restrict__ ei,
                            int* __restrict__ cnt) {
  int e = blockIdx.x * blockDim.x + threadIdx.x;
  if (e < N_EDGES) {
    int r = etype[e];
    int t = ei[N_EDGES + e];            // edge_index[1][e]
    atomicAdd(&cnt[r * N_NODES + t], 1);
  }
}

__global__ void make_inv(const int* __restrict__ cnt, float* __restrict__ inv, int n) {
  int i = blockIdx.x * blockDim.x + threadIdx.x;
  if (i < n) inv[i] = 1.0f / fmaxf((float)cnt[i], 1.0f);
}

__global__ void gather_concat(const int* __restrict__ xidx,
                              const float* __restrict__ semb,
                              const float* __restrict__ cemb,
                              float* __restrict__ xc) {
  int gid = blockIdx.x * blockDim.x + threadIdx.x;
  if (gid >= N_NODES * HID) return;
  int node = gid >> 6;
  int k = gid & 63;
  float v;
  if (k < 32) v = semb[xidx[node * 2 + 0] * 32 + k];
  else        v = cemb[xidx[node * 2 + 1] * 32 + (k - 32)];
  xc[gid] = v;
}

// ---------------- fp32 WMMA matmul: Out[Nx64] = A[Nx64] @ W[64x64] (+bias)(+relu) ----------------
// One wave per 16x16 output tile; K=64 via 16 chained V_WMMA_F32_16X16X4_F32.
// grid = (N/16, 4), block = 32 (single wave32, EXEC all ones, no divergence).

__global__ __launch_bounds__(32) void matmul64_wmma(
    const float* __restrict__ A,     // N x 64
    const float* __restrict__ W,     // 64 x 64 row-major
    const float* __restrict__ bias,  // 64 (may be null)
    float* __restrict__ Out,         // N x 64
    int hasBias, int doRelu) {
  const int m0   = blockIdx.x * 16;
  const int n0   = blockIdx.y * 16;
  const int lane = threadIdx.x;
  const int half = lane >> 4;   // 0: K even pair, 1: K odd pair (per ISA A-frag layout)
  const int l    = lane & 15;

  const float* arow = A + (m0 + l) * HID + 2 * half;       // A[m0+l][2*half + 4*kk ...]
  const float* wcol = W + (2 * half) * HID + n0 + l;       // W[2*half + 4*kk][n0+l]

  v8f c = {};
#pragma unroll
  for (int kk = 0; kk < 16; ++kk) {
    v2f a, b;
    a.x = arow[4 * kk + 0];
    a.y = arow[4 * kk + 1];
    b.x = wcol[(4 * kk) * HID];
    b.y = wcol[(4 * kk) * HID + HID];
    c = __builtin_amdgcn_wmma_f32_16x16x4_f32(
        /*neg_a=*/false, a, /*neg_b=*/false, b,
        /*c_mod=*/(short)0, c, /*reuse_a=*/false, /*reuse_b=*/false);
  }

  const float bv = hasBias ? bias[n0 + l] : 0.0f;
#pragma unroll
  for (int i = 0; i < 8; ++i) {
    float v = c[i] + bv;
    if (doRelu) v = fmaxf(v, 0.0f);
    // C/D layout: VGPR i -> row M = i + 8*half, column N = lane&15
    Out[(m0 + i + 8 * half) * HID + n0 + l] = v;
  }
}

// ---------------- edge scatter (mean aggregation, one relation) ----------------

__global__ void scatter_rel(const float* __restrict__ msg,   // N x 64 (x @ Wr[r])
                            const int* __restrict__ ei,      // 2 x E
                            const int* __restrict__ etype,   // E
                            const float* __restrict__ inv,   // 3 x N
                            int r,
                            float* __restrict__ outx) {      // N x 64 accum
  long long gid = (long long)blockIdx.x * blockDim.x + threadIdx.x;
  int e = (int)(gid >> 6);
  int f = (int)(gid & 63);
  if (e >= N_EDGES) return;
  if (etype[e] != r) return;
  int src = ei[e];
  int tgt = ei[N_EDGES + e];
  float w = inv[r * N_NODES + tgt];
  atomicAdd(&outx[tgt * HID + f], msg[src * HID + f] * w);
}

__global__ void relu_inplace(float* p, int n) {
  int i = blockIdx.x * blockDim.x + threadIdx.x;
  if (i < n) p[i] = fmaxf(p[i], 0.0f);
}

// ---------------- pooling + head ----------------

__global__ void pool_sum(const float* __restrict__ x,
                         const int* __restrict__ batch,
                         float* __restrict__ pooled,   // 512 x 64
                         float* __restrict__ gcnt) {   // 512
  int gid = blockIdx.x * blockDim.x + threadIdx.x;
  if (gid >= N_NODES * HID) return;
  int node = gid >> 6;
  int f = gid & 63;
  int g = batch[node];
  atomicAdd(&pooled[g * HID + f], x[gid]);
  if (f == 0) atomicAdd(&gcnt[g], 1.0f);
}

__global__ void head_kernel(const float* __restrict__ pooled,
                            const float* __restrict__ gcnt,
                            const float* __restrict__ Wout,  // 64 x 2
                            const float* __restrict__ bout,  // 2
                            float* __restrict__ out) {       // 512 x 2
  int t = blockIdx.x * blockDim.x + threadIdx.x;
  if (t >= N_GRAPHS * 2) return;
  int g = t >> 1;
  int c = t & 1;
  float invc = 1.0f / fmaxf(gcnt[g], 1.0f);
  float s = 0.0f;
#pragma unroll
  for (int h = 0; h < HID; ++h)
    s += pooled[g * HID + h] * invc * Wout[h * 2 + c];
  out[t] = s + bout[c];
}

// ---------------- launch ----------------

extern "C" void kernel_launch(void* const* d_in, const int* in_sizes, int n_in,
                              void* d_out, int out_size, void* d_ws, size_t ws_size,
                              hipStream_t stream) {
  const int*   x_idx  = (const int*)d_in[0];
  const int*   eindex = (const int*)d_in[1];
  const int*   etype  = (const int*)d_in[2];
  const int*   batch  = (const int*)d_in[3];
  const float* semb   = (const float*)d_in[4];
  const float* cemb   = (const float*)d_in[5];
  const float* W_in   = (const float*)d_in[6];
  const float* b_in   = (const float*)d_in[7];
  const float* Wr[3]   = { (const float*)d_in[8],  (const float*)d_in[11], (const float*)d_in[14] };
  const float* root[3] = { (const float*)d_in[9],  (const float*)d_in[12], (const float*)d_in[15] };
  const float* bl[3]   = { (const float*)d_in[10], (const float*)d_in[13], (const float*)d_in[16] };
  const float* W_out  = (const float*)d_in[17];
  const float* b_out  = (const float*)d_in[18];
  float* out = (float*)d_out;

  // workspace carve-up (bytes)
  char* ws = (char*)d_ws;
  size_t off = 0;
  float* bufA   = (float*)(ws + off); off += (size_t)N_NODES * HID * 4;       // 25.6 MB
  float* bufB   = (float*)(ws + off); off += (size_t)N_NODES * HID * 4;       // 25.6 MB
  float* msg    = (float*)(ws + off); off += (size_t)N_NODES * HID * 4;       // 25.6 MB
  int*   cnt    = (int*)  (ws + off); off += (size_t)NREL * N_NODES * 4;      // 1.2 MB
  float* inv    = (float*)(ws + off); off += (size_t)NREL * N_NODES * 4;      // 1.2 MB
  float* pooled = (float*)(ws + off); off += (size_t)N_GRAPHS * HID * 4;      // 128 KB
  float* gcnt   = (float*)(ws + off); off += (size_t)N_GRAPHS * 4;

  const int NH = N_NODES * HID;
  const dim3 mmGrid(N_NODES / 16, HID / 16);   // 100000 = 6250*16 exactly

  // ---- per-(relation,target) edge counts -> inverse means (layer-invariant) ----
  zero_u32<<<(NREL * N_NODES + 255) / 256, 256, 0, stream>>>((unsigned int*)cnt, NREL * N_NODES);
  count_edges<<<(N_EDGES + 255) / 256, 256, 0, stream>>>(etype, eindex, cnt);
  make_inv<<<(NREL * N_NODES + 255) / 256, 256, 0, stream>>>(cnt, inv, NREL * N_NODES);

  // ---- input layer: xc -> x0 = relu(xc @ W_in + b_in) ----
  gather_concat<<<(NH + 255) / 256, 256, 0, stream>>>(x_idx, semb, cemb, bufB);
  matmul64_wmma<<<mmGrid, 32, 0, stream>>>(bufB, W_in, b_in, bufA, 1, 1);

  float* x  = bufA;
  float* nx = bufB;

  // ---- 3 R-GCN layers ----
  for (int layer = 0; layer < 3; ++layer) {
    // nx = x @ root + b  (self term; overwrites previous contents)
    matmul64_wmma<<<mmGrid, 32, 0, stream>>>(x, root[layer], bl[layer], nx, 1, 0);
    for (int r = 0; r < NREL; ++r) {
      // msg = x @ Wr[r]
      matmul64_wmma<<<mmGrid, 32, 0, stream>>>(x, Wr[layer] + (size_t)r * HID * HID,
                                               nullptr, msg, 0, 0);
      // nx[tgt] += msg[src] * inv_cnt[r][tgt]   (atomics, L2-resident)
      long long total = (long long)N_EDGES * HID;
      scatter_rel<<<(unsigned)((total + 255) / 256), 256, 0, stream>>>(
          msg, eindex, etype, inv, r, nx);
    }
    relu_inplace<<<(NH + 255) / 256, 256, 0, stream>>>(nx, NH);
    float* t = x; x = nx; nx = t;
  }

  // ---- mean pool per graph + linear head ----
  zero_u32<<<(N_GRAPHS * HID + 255) / 256, 256, 0, stream>>>((unsigned int*)pooled, N_GRAPHS * HID);
  zero_u32<<<(N_GRAPHS + 255) / 256, 256, 0, stream>>>((unsigned int*)gcnt, N_GRAPHS);
  pool_sum<<<(NH + 255) / 256, 256, 0, stream>>>(x, batch, pooled, gcnt);
  head_kernel<<<(N_GRAPHS * 2 + 255) / 256, 256, 0, stream>>>(pooled, gcnt, W_out, b_out, out);
}